// VQVAE_Encoder_86998857548102
// MI455X (gfx1250) — compile-verified
//
#include <hip/hip_runtime.h>
#include <hip/hip_bf16.h>
#include <math.h>

typedef __attribute__((ext_vector_type(2))) float v2f;
typedef __attribute__((ext_vector_type(8))) float v8f;

// D(16x16,f32) += A(16x4,f32) * B(4x16,f32), full-wave op (wave32)
static __device__ __forceinline__ v8f wmma4(v2f a, v2f b, v8f c) {
  return __builtin_amdgcn_wmma_f32_16x16x4_f32(
      /*neg_a=*/false, a, /*neg_b=*/false, b,
      /*c_mod=*/(short)0, c, /*reuse_a=*/false, /*reuse_b=*/false);
}

// ---------------------------------------------------------------------------
// 1D conv as GEMM, compile-time specialized. One wave owns a 16(co) x 32(t)
// tile: the A (weight) fragment is reused by two WMMAs per reduction step.
// Boundary handling is branch-free (clamped address * 0/1 mask), so EXEC
// stays all-1s for the entire kernel (WMMA requirement, zero saveexec).
// ---------------------------------------------------------------------------
template <int KW, int STRIDE, int PAD, int CIN,
          bool RELU_IN, bool RELU_OUT, bool TRANSPOSE, bool HAS_BIAS, bool HAS_ADD>
__global__ __launch_bounds__(256) void conv1d_wmma_kernel(
    const float* __restrict__ xin, const float* __restrict__ w,
    const float* __restrict__ bias, const float* __restrict__ addsrc,
    float* __restrict__ y, int Tin, int Cout, int Tout)
{
  const int wave = (int)((blockIdx.x * blockDim.x + threadIdx.x) >> 5);
  const int lane = (int)(threadIdx.x & 31);
  const int tilesT = Tout >> 5;            // 32-wide t tiles
  const int tilesC = Cout >> 4;
  // grid sized exactly: no bounds check needed
  const int perB = tilesC * tilesT;
  int b   = wave / perB;
  int rem = wave - b * perB;
  int co0 = (rem / tilesT) << 4;
  int t0  = (rem % tilesT) << 5;
  const int m = lane & 15, half = lane >> 4;

  v8f acc0 = {}, acc1 = {};
  const float* xb = xin + (size_t)b * CIN * Tin;
  const float* wr = w + (size_t)(co0 + m) * CIN * KW;   // A row for this lane

#pragma unroll
  for (int k = 0; k < KW; ++k) {
    const int pos0 = STRIDE * (t0 + m) - PAD + k;       // B columns (two subtiles)
    const int pos1 = pos0 + 16 * STRIDE;
    const float m0 = (pos0 >= 0 && pos0 < Tin) ? 1.0f : 0.0f;
    const float m1 = (pos1 >= 0 && pos1 < Tin) ? 1.0f : 0.0f;
    const int p0 = min(max(pos0, 0), Tin - 1);
    const int p1 = min(max(pos1, 0), Tin - 1);
#pragma unroll 4
    for (int ci = 0; ci < CIN; ci += 4) {
      const int cia = ci + 2 * half;                    // K-slot this lane supplies
      v2f a;
      a.x = wr[(size_t)cia * KW + k];
      a.y = wr[(size_t)(cia + 1) * KW + k];
      const float* xc0 = xb + (size_t)cia * Tin;
      const float* xc1 = xb + (size_t)(cia + 1) * Tin;
      float u0 = xc0[p0], u1 = xc1[p0];                 // subtile 0
      float s0 = xc0[p1], s1 = xc1[p1];                 // subtile 1
      if (RELU_IN) {
        u0 = fmaxf(u0, 0.0f); u1 = fmaxf(u1, 0.0f);
        s0 = fmaxf(s0, 0.0f); s1 = fmaxf(s1, 0.0f);
      }
      v2f b0; b0.x = u0 * m0; b0.y = u1 * m0;
      v2f b1; b1.x = s0 * m1; b1.y = s1 * m1;
      acc0 = wmma4(a, b0, acc0);                        // A reused by both WMMAs
      acc1 = wmma4(a, b1, acc1);
    }
  }

  const int t_a = t0 + m, t_b = t0 + 16 + m;
#pragma unroll
  for (int v = 0; v < 8; ++v) {
    const int co = co0 + v + 8 * half;                  // C/D layout: M = v + 8*half
    const float bv = HAS_BIAS ? bias[co] : 0.0f;
    float va = acc0[v] + bv;
    float vb = acc1[v] + bv;
    if (HAS_ADD) {
      va += addsrc[((size_t)b * Cout + co) * Tout + t_a];
      vb += addsrc[((size_t)b * Cout + co) * Tout + t_b];
    }
    if (RELU_OUT) { va = fmaxf(va, 0.0f); vb = fmaxf(vb, 0.0f); }
    if (TRANSPOSE) {
      y[((size_t)b * Tout + t_a) * Cout + co] = va;     // (B,T,C)
      y[((size_t)b * Tout + t_b) * Cout + co] = vb;
    } else {
      y[((size_t)b * Cout + co) * Tout + t_a] = va;     // (B,C,T)
      y[((size_t)b * Cout + co) * Tout + t_b] = vb;
    }
  }
}

// ---------------------------------------------------------------------------
// Init: codebook row norms, zero histogram + sq-error accumulator.
// ---------------------------------------------------------------------------
__global__ void vq_init_kernel(const float* __restrict__ cb, float* __restrict__ cnorm,
                               float* __restrict__ hist, float* __restrict__ sqsum)
{
  int t = (int)(blockIdx.x * blockDim.x + threadIdx.x);
  if (t < 512) {
    float s = 0.0f;
    for (int d = 0; d < 64; ++d) { float v = cb[(size_t)t * 64 + d]; s += v * v; }
    cnorm[t] = s;
    hist[t]  = 0.0f;
  }
  if (t == 0) sqsum[0] = 0.0f;
}

// ---------------------------------------------------------------------------
// VQ: each wave owns 16 z-rows (A held in 32 VGPRs), streams the 512 codebook
// entries through 16-wide WMMA tiles, tracks running argmin in the C-matrix
// register layout, shuffle-reduces, then writes z_q_st / one-hot / indices /
// histogram / squared-error.
// ---------------------------------------------------------------------------
__global__ __launch_bounds__(256) void vq_kernel(
    const float* __restrict__ z,      // (N,64) = z_e, row-major
    const float* __restrict__ cb,     // (512,64)
    const float* __restrict__ cnorm,  // (512)
    float* __restrict__ zq,           // (N,64)  z_q_st
    float* __restrict__ enc,          // (N,512) one-hot
    float* __restrict__ oidx,         // (N)     indices (as float)
    float* __restrict__ hist,         // (512)
    float* __restrict__ sqsum,        // (1)
    int N)
{
  const int wave = (int)((blockIdx.x * blockDim.x + threadIdx.x) >> 5);
  const int lane = (int)(threadIdx.x & 31);
  if (wave >= (N >> 4)) return;              // wave-uniform
  const int n0 = wave << 4;
  const int m = lane & 15, half = lane >> 4;
  const int D = 64, K = 512;

  v2f A[16];
  float zp = 0.0f;
#pragma unroll
  for (int s = 0; s < 16; ++s) {
    A[s] = *(const v2f*)(z + (size_t)(n0 + m) * D + 4 * s + 2 * half);
    zp += A[s].x * A[s].x + A[s].y * A[s].y;
  }
  zp += __shfl_xor(zp, 16);                  // full |z_row|^2 on every lane
  float znV[8];
#pragma unroll
  for (int v = 0; v < 8; ++v) znV[v] = __shfl(zp, v + 8 * half);

  float best[8]; int bidx[8];
#pragma unroll
  for (int v = 0; v < 8; ++v) { best[v] = 3.4e38f; bidx[v] = 0; }

  for (int kb = 0; kb < K; kb += 16) {
    v8f acc = {};
#pragma unroll
    for (int s = 0; s < 16; ++s) {
      v2f bv = *(const v2f*)(cb + (size_t)(kb + m) * D + 4 * s + 2 * half);
      acc = wmma4(A[s], bv, acc);            // acc[v] = z[n0+v+8h] . cb[kb+m']
    }
    float cn = cnorm[kb + m];
    int kk = kb + m;
#pragma unroll
    for (int v = 0; v < 8; ++v) {
      float d = znV[v] + cn - 2.0f * acc[v];
      if (d < best[v]) { best[v] = d; bidx[v] = kk; }  // ascending k -> first-tie wins
    }
  }

#pragma unroll
  for (int v = 0; v < 8; ++v) {
#pragma unroll
    for (int off = 8; off >= 1; off >>= 1) {
      float od = __shfl_xor(best[v], off);
      int   oi = __shfl_xor(bidx[v], off);
      if (od < best[v] || (od == best[v] && oi < bidx[v])) { best[v] = od; bidx[v] = oi; }
    }
  }
  int rowIdx[16];
#pragma unroll
  for (int r = 0; r < 16; ++r)
    rowIdx[r] = __shfl(bidx[r & 7], (r >> 3) << 4);    // row r lives in half r>>3

  float lsq = 0.0f;
#pragma unroll 1
  for (int r = 0; r < 16; ++r) {
    int n = n0 + r;
    int kwin = rowIdx[r];
    float c0 = cb[(size_t)kwin * D + lane];
    float c1 = cb[(size_t)kwin * D + 32 + lane];
    float e0 = z[(size_t)n * D + lane];
    float e1 = z[(size_t)n * D + 32 + lane];
    zq[(size_t)n * D + lane]      = e0 + (c0 - e0);    // z_q_st = z_e + (c - z_e)
    zq[(size_t)n * D + 32 + lane] = e1 + (c1 - e1);
    float d0 = c0 - e0, d1 = c1 - e1;
    lsq += d0 * d0 + d1 * d1;
#pragma unroll
    for (int j = 0; j < 16; ++j) {
      int col = j * 32 + lane;
      enc[(size_t)n * 512 + col] = (col == kwin) ? 1.0f : 0.0f;
    }
    if (lane == 0) {
      oidx[n] = (float)kwin;
      atomicAdd(&hist[kwin], 1.0f);
    }
  }
#pragma unroll
  for (int off = 16; off >= 1; off >>= 1) lsq += __shfl_xor(lsq, off);
  if (lane == 0) atomicAdd(sqsum, lsq);
}

// ---------------------------------------------------------------------------
// Scalars: perplexity = exp(-sum p log(p+1e-10)); loss = 1.25*sqsum/(N*D)
// ---------------------------------------------------------------------------
__global__ void vq_finalize_kernel(const float* __restrict__ hist,
                                   const float* __restrict__ sqsum,
                                   float* __restrict__ out_loss,
                                   float* __restrict__ out_perp, int N, int D)
{
  __shared__ float red[512];
  int t = (int)threadIdx.x;
  float p = hist[t] / (float)N;
  red[t] = p * logf(p + 1e-10f);
  __syncthreads();
  for (int s = 256; s > 0; s >>= 1) { if (t < s) red[t] += red[t + s]; __syncthreads(); }
  if (t == 0) {
    out_perp[0] = expf(-red[0]);
    out_loss[0] = 1.25f * sqsum[0] / ((float)N * (float)D);  // (1+BETA)*mse
  }
}

// ---------------------------------------------------------------------------
template <int KW, int S, int P, int CIN,
          bool RI, bool RO, bool TR, bool HB, bool HA>
static void run_conv(const float* x, const float* w, const float* bias,
                     const float* add, float* y,
                     int Tin, int Cout, int Tout, hipStream_t stream) {
  int waves = 64 * (Cout / 16) * (Tout / 32);   // B = 64
  conv1d_wmma_kernel<KW, S, P, CIN, RI, RO, TR, HB, HA>
      <<<dim3(waves / 8), dim3(256), 0, stream>>>(x, w, bias, add, y, Tin, Cout, Tout);
}

extern "C" void kernel_launch(void* const* d_in, const int* in_sizes, int n_in,
                              void* d_out, int out_size, void* d_ws, size_t ws_size,
                              hipStream_t stream) {
  const float* x     = (const float*)d_in[0];
  const float* w1    = (const float*)d_in[1];
  const float* b1    = (const float*)d_in[2];
  const float* w2    = (const float*)d_in[3];
  const float* b2    = (const float*)d_in[4];
  const float* w3    = (const float*)d_in[5];
  const float* b3    = (const float*)d_in[6];
  const float* r0w1  = (const float*)d_in[7];
  const float* r0w2  = (const float*)d_in[8];
  const float* r1w1  = (const float*)d_in[9];
  const float* r1w2  = (const float*)d_in[10];
  const float* w_out = (const float*)d_in[11];
  const float* b_out = (const float*)d_in[12];
  const float* wpq   = (const float*)d_in[13];
  const float* bpq   = (const float*)d_in[14];
  const float* cbk   = (const float*)d_in[15];

  float* out = (float*)d_out;
  float* ws  = (float*)d_ws;

  // workspace layout (floats)
  float* R0    = ws;                  // 8388608  (64x128x1024 / 64x64x2048)
  float* R1    = ws + 8388608;        // 8388608
  float* R2    = ws + 16777216;       // 4194304  (64x64x1024)
  float* hist  = ws + 20971520;       // 512
  float* cnorm = ws + 20972032;       // 512
  float* sq    = ws + 20972544;       // 1

  // d_out layout (tuple concatenated flat, floats)
  float* o_loss = out;                // 1
  float* o_zq   = out + 1;            // 4194304
  float* o_perp = out + 4194305;      // 1
  float* o_ze   = out + 4194306;      // 4194304 (B,T',D) row-major == VQ input
  float* o_enc  = out + 8388610;      // 33554432
  float* o_idx  = out + 41943042;     // 65536

  // encoder                                   KW S  P  CIN  RI RO TR HB HA
  run_conv<4, 2, 1,  64, false, true,  false, true,  false>(x,  w1, b1, nullptr, R0, 4096,  64, 2048, stream);
  run_conv<4, 2, 1,  64, false, true,  false, true,  false>(R0, w2, b2, nullptr, R1, 2048, 128, 1024, stream);
  run_conv<3, 1, 1, 128, false, false, false, true,  false>(R1, w3, b3, nullptr, R0, 1024, 128, 1024, stream);
  // residual block 0: R1 = R0 + conv1x1(relu(conv3(relu(R0))))
  run_conv<3, 1, 1, 128, true,  true,  false, false, false>(R0, r0w1, nullptr, nullptr, R2, 1024,  64, 1024, stream);
  run_conv<1, 1, 0,  64, false, false, false, false, true >(R2, r0w2, nullptr, R0,      R1, 1024, 128, 1024, stream);
  // residual block 1: R0 = R1 + conv1x1(relu(conv3(relu(R1))))
  run_conv<3, 1, 1, 128, true,  true,  false, false, false>(R1, r1w1, nullptr, nullptr, R2, 1024,  64, 1024, stream);
  run_conv<1, 1, 0,  64, false, false, false, false, true >(R2, r1w2, nullptr, R1,      R0, 1024, 128, 1024, stream);
  // head: relu -> w_out -> wpq (z_e written directly in (B,T,D) layout)
  run_conv<1, 1, 0, 128, true,  false, false, true,  false>(R0, w_out, b_out, nullptr, R2,  1024, 64, 1024, stream);
  run_conv<1, 1, 0,  64, false, false, true,  true,  false>(R2, wpq,   bpq,   nullptr, o_ze, 1024, 64, 1024, stream);

  // vector quantization
  const int N = 64 * 1024;            // 65536 rows
  vq_init_kernel<<<dim3(1), dim3(512), 0, stream>>>(cbk, cnorm, hist, sq);
  vq_kernel<<<dim3((N / 16) / 8), dim3(256), 0, stream>>>(
      o_ze, cbk, cnorm, o_zq, o_enc, o_idx, hist, sq, N);
  vq_finalize_kernel<<<dim3(1), dim3(512), 0, stream>>>(hist, sq, o_loss, o_perp, N, 64);
}